// Coords2RMSD_54443005444629
// MI455X (gfx1250) — compile-verified
//
#include <hip/hip_runtime.h>

typedef float v2f __attribute__((ext_vector_type(2)));
typedef float v8f __attribute__((ext_vector_type(8)));

#define BATCH            8192
#define NATOMS           512
#define WAVES_PER_BLOCK  8
#define THREADS          (WAVES_PER_BLOCK * 32)
#define CHUNK            32          // atoms staged per wave per chunk
#define SLOT             20          // float words per atom slot (>=16; 20 -> conflict-free reads)
#define NCHUNK           (NATOMS / CHUNK)

// Gram-matrix formulation of Kabsch RMSD:
//   M = [sx sy sz dx dy dz 1] * mask  (N x 7, padded to 16 features)
//   G = M^T M  accumulated with V_WMMA_F32_16X16X4_F32 over K = atoms.
// G contains Sum(s d^T), Sum(s), Sum(d), Sum|s|^2, Sum|d|^2, n.
__global__ __launch_bounds__(THREADS)
void kabsch_rmsd_wmma(const float* __restrict__ inp,
                      const float* __restrict__ tgt,
                      const int*   __restrict__ num_atoms,
                      float*       __restrict__ out)
{
    __shared__ float lds[WAVES_PER_BLOCK * CHUNK * SLOT];

    const int tid  = threadIdx.x;
    const int lane = tid & 31;
    const int wave = tid >> 5;
    const int b    = blockIdx.x * WAVES_PER_BLOCK + wave;

    float* wbase = lds + wave * CHUNK * SLOT;

    // Zero the staging slab once: feature words 7..19 of every slot must read 0.
    for (int i = tid; i < WAVES_PER_BLOCK * CHUNK * SLOT; i += THREADS) lds[i] = 0.0f;
    __syncthreads();

    const int   na  = num_atoms[b];
    const float naf = (float)na;

    const float* src = inp + (size_t)b * (3 * NATOMS);
    const float* dst = tgt + (size_t)b * (3 * NATOMS);

    v8f acc = {};                 // 16x16 fp32 accumulator = Gram matrix G

    const int feat = lane & 15;   // feature row/col index for WMMA fragment
    const int hcol = lane >> 4;   // 0 -> K slots {0,1}; 1 -> K slots {2,3}

    for (int c = 0; c < NCHUNK; ++c) {
        // ---- stage: lane stages atom (c*CHUNK + lane), coalesced float3 loads
        const int atom = c * CHUNK + lane;
        const float m  = (atom < na) ? 1.0f : 0.0f;
        const float* s3 = src + 3 * atom;
        const float* d3 = dst + 3 * atom;
        const float sx = m * s3[0], sy = m * s3[1], sz = m * s3[2];
        const float dx = m * d3[0], dy = m * d3[1], dz = m * d3[2];

        float* slot = wbase + lane * SLOT;          // 80-byte stride, 16B aligned
        *(float4*)(slot)     = make_float4(sx, sy, sz, dx);
        *(float2*)(slot + 4) = make_float2(dy, dz);
        slot[6] = m;
        __syncthreads();

        // ---- 8 WMMA steps, K = 4 atoms each. A and B fragments are identical
        // for M^T M under the documented f32 A/B lane layouts.
        #pragma unroll
        for (int t = 0; t < 8; ++t) {
            const int k0 = 4 * t + 2 * hcol;        // K slots {0,2} in .x, {1,3} in .y
            v2f a;
            a.x = wbase[(k0 + 0) * SLOT + feat];
            a.y = wbase[(k0 + 1) * SLOT + feat];
            acc = __builtin_amdgcn_wmma_f32_16x16x4_f32(
                      false, a, false, a, (short)0, acc, false, false);
        }
        __syncthreads();
    }

    // ---- extract Gram entries: row r of G = acc[r], col = lane (rows 0..7 in lanes 0..15)
    float Sm[3][3], ss[3], sd[3];
    #pragma unroll
    for (int i = 0; i < 3; ++i) {
        #pragma unroll
        for (int j = 0; j < 3; ++j)
            Sm[i][j] = __shfl(acc[i], 3 + j, 32);   // Sum s_i d_j
        ss[i] = __shfl(acc[i],     6, 32);          // Sum s_i
        sd[i] = __shfl(acc[3 + i], 6, 32);          // Sum d_i
    }
    const float ssq = __shfl(acc[0], 0, 32) + __shfl(acc[1], 1, 32) + __shfl(acc[2], 2, 32);
    const float dsq = __shfl(acc[3], 3, 32) + __shfl(acc[4], 4, 32) + __shfl(acc[5], 5, 32);

    const float inv_n = 1.0f / naf;

    // centered correlation matrix R and centered squared norms
    float R[3][3];
    #pragma unroll
    for (int i = 0; i < 3; ++i)
        #pragma unroll
        for (int j = 0; j < 3; ++j)
            R[i][j] = Sm[i][j] - ss[i] * sd[j] * inv_n;

    const float sq = ssq + dsq
                   - (ss[0]*ss[0] + ss[1]*ss[1] + ss[2]*ss[2]) * inv_n
                   - (sd[0]*sd[0] + sd[1]*sd[1] + sd[2]*sd[2]) * inv_n;

    const float detR =
          R[0][0] * (R[1][1]*R[2][2] - R[1][2]*R[2][1])
        - R[0][1] * (R[1][0]*R[2][2] - R[1][2]*R[2][0])
        + R[0][2] * (R[1][0]*R[2][1] - R[1][1]*R[2][0]);

    // A = R^T R (symmetric PSD); its eigenvalues are squared singular values of R
    const float A00 = R[0][0]*R[0][0] + R[1][0]*R[1][0] + R[2][0]*R[2][0];
    const float A11 = R[0][1]*R[0][1] + R[1][1]*R[1][1] + R[2][1]*R[2][1];
    const float A22 = R[0][2]*R[0][2] + R[1][2]*R[1][2] + R[2][2]*R[2][2];
    const float A01 = R[0][0]*R[0][1] + R[1][0]*R[1][1] + R[2][0]*R[2][1];
    const float A02 = R[0][0]*R[0][2] + R[1][0]*R[1][2] + R[2][0]*R[2][2];
    const float A12 = R[0][1]*R[0][2] + R[1][1]*R[1][2] + R[2][1]*R[2][2];

    // analytic symmetric-3x3 eigenvalues (trigonometric method), e1 >= e2 >= e3
    const float p1 = A01*A01 + A02*A02 + A12*A12;
    const float q  = (A00 + A11 + A22) * (1.0f / 3.0f);
    const float p2 = (A00-q)*(A00-q) + (A11-q)*(A11-q) + (A22-q)*(A22-q) + 2.0f*p1;
    const float p    = sqrtf(fmaxf(p2, 1e-30f) * (1.0f / 6.0f));
    const float invp = 1.0f / p;
    const float B00 = (A00 - q) * invp, B11 = (A11 - q) * invp, B22 = (A22 - q) * invp;
    const float B01 = A01 * invp, B02 = A02 * invp, B12 = A12 * invp;
    const float detB = B00*(B11*B22 - B12*B12)
                     - B01*(B01*B22 - B12*B02)
                     + B02*(B01*B12 - B11*B02);
    const float rr  = fminf(fmaxf(0.5f * detB, -1.0f), 1.0f);
    const float phi = acosf(rr) * (1.0f / 3.0f);
    const float e1  = q + 2.0f * p * cosf(phi);
    const float e3  = q + 2.0f * p * cosf(phi + 2.0943951023931953f); // +2pi/3
    const float e2  = 3.0f * q - e1 - e3;

    const float s0 = sqrtf(fmaxf(e1, 0.0f));
    const float s1 = sqrtf(fmaxf(e2, 0.0f));
    const float s2 = sqrtf(fmaxf(e3, 0.0f));

    const float chi = (detR > 0.0f) ? 1.0f : ((detR < 0.0f) ? -1.0f : 0.0f);
    const float trace_opt = s0 + s1 + chi * s2;

    const float msd  = (sq - 2.0f * trace_opt) * inv_n;
    const float rmsd = sqrtf(fmaxf(msd, 0.0f));

    if (lane == 0) out[b] = rmsd;
}

extern "C" void kernel_launch(void* const* d_in, const int* in_sizes, int n_in,
                              void* d_out, int out_size, void* d_ws, size_t ws_size,
                              hipStream_t stream) {
    const float* inp = (const float*)d_in[0];
    const float* tgt = (const float*)d_in[1];
    const int*   na  = (const int*)d_in[2];
    float*       out = (float*)d_out;

    dim3 grid(BATCH / WAVES_PER_BLOCK);   // 1024 blocks x 8 waves = 8192 waves, 1 per structure
    kabsch_rmsd_wmma<<<grid, THREADS, 0, stream>>>(inp, tgt, na, out);
}